// RadiusInteractionGraph_65876208386291
// MI455X (gfx1250) — compile-verified
//
#include <hip/hip_runtime.h>
#include <hip/hip_bf16.h>
#include <math.h>

// ---- problem constants (match reference) ----
#define N_ATOMS 8192
#define KNBR    32
#define CUT2    100.0f   // CUTOFF^2

typedef __attribute__((ext_vector_type(2))) float v2f;
typedef __attribute__((ext_vector_type(8))) float v8f;

__device__ __forceinline__ int lowerBound(const int* __restrict__ a, int n, int key) {
    int lo = 0, hi = n;
    while (lo < hi) { int mid = (lo + hi) >> 1; if (a[mid] < key) lo = mid + 1; else hi = mid; }
    return lo;
}
__device__ __forceinline__ int upperBound(const int* __restrict__ a, int n, int key) {
    int lo = 0, hi = n;
    while (lo < hi) { int mid = (lo + hi) >> 1; if (a[mid] <= key) lo = mid + 1; else hi = mid; }
    return lo;
}

// One wave per 32-row tile. Squared-distance tiles via V_WMMA_F32_16X16X4_F32:
//   A_i = (x, y, z, |p_i|^2),  B_j = (-2x, -2y, -2z, 1)
//   => D[i][j] = -2 p_i.p_j + |p_i|^2 ; then + |p_j|^2 (per-lane column norm).
// Two WMMAs share one B operand (rows r0..r0+15 and r0+16..r0+31).
__global__ __launch_bounds__(32)
void RadiusInteractionGraph_kernel(const float* __restrict__ pos,
                                   const int*   __restrict__ batch,
                                   float*       __restrict__ out) {
    __shared__ float sh_sq[32][17];   // [tile row][tile col], padded
    __shared__ int   sh_cb[16];

    const int lane = threadIdx.x;     // 0..31
    const int half = lane >> 4;       // K-group select in A/B operand layout
    const int m    = lane & 15;       // row/col index within 16-wide operand
    const int r0   = blockIdx.x * 32;

    // ---- A operands: rows r0..r0+15 (A0) and r0+16..r0+31 (A1) ----
    const int   ra0 = r0 + m;
    const int   ra1 = r0 + 16 + m;
    const float a0x = pos[3 * ra0 + 0], a0y = pos[3 * ra0 + 1], a0z = pos[3 * ra0 + 2];
    const float a1x = pos[3 * ra1 + 0], a1y = pos[3 * ra1 + 1], a1z = pos[3 * ra1 + 2];
    const float a0n = a0x * a0x + a0y * a0y + a0z * a0z;
    const float a1n = a1x * a1x + a1y * a1y + a1z * a1z;
    v2f A0, A1;
    A0.x = half ? a0z : a0x;  A0.y = half ? a0n : a0y;
    A1.x = half ? a1z : a1x;  A1.y = half ? a1n : a1y;

    // lane owns row r0+lane in the selection phase
    const int   myRow = r0 + lane;
    const float ox = half ? a1x : a0x;
    const float oy = half ? a1y : a0y;
    const float oz = half ? a1z : a0z;
    const int   myBatch = batch[myRow];

    // ---- uniform candidate column range (batch is sorted) ----
    const int bFirst   = batch[r0];
    const int bLast    = batch[r0 + 31];
    const int colStart = lowerBound(batch, N_ATOMS, bFirst);
    const int colEnd   = upperBound(batch, N_ATOMS, bLast);
    const int tile0    = colStart & ~15;

    // ---- per-lane sorted top-K (ascending sq) in registers ----
    float bq[KNBR];
    int   bi[KNBR];
#pragma unroll
    for (int k = 0; k < KNBR; ++k) { bq[k] = INFINITY; bi[k] = myRow; }

    for (int c0 = tile0; c0 < colEnd; c0 += 16) {
        // ---- B operand + per-column metadata ----
        const int   c  = c0 + m;
        const int   cc = (c < N_ATOMS) ? c : (N_ATOMS - 1);
        const float bx = pos[3 * cc + 0];
        const float by = pos[3 * cc + 1];
        const float bz = pos[3 * cc + 2];
        const float bn = bx * bx + by * by + bz * bz;
        const int   cb = (c < N_ATOMS) ? batch[cc] : -1;
        v2f B;
        B.x = half ? (-2.0f * bz) : (-2.0f * bx);
        B.y = half ? 1.0f         : (-2.0f * by);

        v8f acc0 = {};
        acc0 = __builtin_amdgcn_wmma_f32_16x16x4_f32(false, A0, false, B,
                                                     (short)0, acc0, false, false);
        v8f acc1 = {};
        acc1 = __builtin_amdgcn_wmma_f32_16x16x4_f32(false, A1, false, B,
                                                     (short)0, acc1, false, false);

        // D layout: VGPR r, this lane -> tile row (r + 8*half), col m.
        // Add column norm and store transposed through LDS.
#pragma unroll
        for (int r = 0; r < 8; ++r) {
            sh_sq[r + 8 * half][m]      = acc0[r] + bn;
            sh_sq[16 + r + 8 * half][m] = acc1[r] + bn;
        }
        if (half == 0) sh_cb[m] = cb;
        __syncthreads();

        // ---- branchless streaming top-K merge: every lane owns one row ----
#pragma unroll 1
        for (int n = 0; n < 16; ++n) {
            const int  cj = c0 + n;
            float      sq = sh_sq[lane][n];
            const bool ok = (sh_cb[n] == myBatch) && (cj != myRow) && (sq <= CUT2);
            sq = ok ? sq : INFINITY;    // invalid candidate -> merge is a no-op
#pragma unroll
            for (int p = KNBR - 1; p >= 0; --p) {
                const float prev_q = (p > 0) ? bq[p - 1] : -INFINITY;
                const int   prev_i = (p > 0) ? bi[p - 1] : cj;
                const bool  keep   = (bq[p] <= sq);   // at/below insertion point (ties keep old)
                const bool  shift  = (prev_q > sq);   // above insertion point
                const float nq = keep ? bq[p] : (shift ? prev_q : sq);
                const int   ni = keep ? bi[p] : (shift ? prev_i : cj);
                bq[p] = nq; bi[p] = ni;
            }
        }
        __syncthreads();
    }

    // ---- emit: edge_index(row), edge_index(col), edge_weight, edge_mask ----
    {
        const size_t NE    = (size_t)N_ATOMS * KNBR;
        float* o_row = out;
        float* o_col = out + NE;
        float* o_w   = out + 2 * NE;
        float* o_msk = out + 3 * NE;
#pragma unroll 1
        for (int k = 0; k < KNBR; ++k) {
            const size_t e    = (size_t)myRow * KNBR + k;
            const bool   real = (bq[k] < INFINITY);
            const int    cj   = bi[k];
            float w = 0.0f;
            if (real) {  // exact recompute from gathered positions (matches reference)
                const float dx = ox - pos[3 * cj + 0];
                const float dy = oy - pos[3 * cj + 1];
                const float dz = oz - pos[3 * cj + 2];
                const float s2 = dx * dx + dy * dy + dz * dz;
                w = sqrtf(fmaxf(s2, 1e-12f));
            }
            o_row[e] = (float)myRow;
            o_col[e] = (float)cj;
            o_w[e]   = w;
            o_msk[e] = real ? 1.0f : 0.0f;
        }
    }
}

extern "C" void kernel_launch(void* const* d_in, const int* in_sizes, int n_in,
                              void* d_out, int out_size, void* d_ws, size_t ws_size,
                              hipStream_t stream) {
    (void)in_sizes; (void)n_in; (void)out_size; (void)d_ws; (void)ws_size;
    const float* pos   = (const float*)d_in[0];
    const int*   batch = (const int*)d_in[1];
    float*       out   = (float*)d_out;

    dim3 grid(N_ATOMS / 32);   // 256 tiles of 32 rows, one wave each
    dim3 block(32);
    RadiusInteractionGraph_kernel<<<grid, block, 0, stream>>>(pos, batch, out);
}